// Aggregationfeature_80161269612654
// MI455X (gfx1250) — compile-verified
//
#include <hip/hip_runtime.h>
#include <hip/hip_bf16.h>

typedef __attribute__((ext_vector_type(2))) float v2f;
typedef __attribute__((ext_vector_type(8))) float v8f;

#define NN     38000
#define RR     8
#define EE     200000
#define NROWS  60000
#define FF     19
#define HID    128
#define INDIM  64
#define NCLS   10

// ---------------- degree / norm ----------------
__global__ void count_deg_kernel(const int* __restrict__ src,
                                 const int* __restrict__ dst,
                                 float* __restrict__ deg_s,
                                 float* __restrict__ deg_d) {
    int i = blockIdx.x * blockDim.x + threadIdx.x;
    if (i >= RR * EE) return;
    int r = i / EE;
    atomicAdd(&deg_s[(size_t)r * NN + src[i]], 1.0f);
    atomicAdd(&deg_d[(size_t)r * NN + dst[i]], 1.0f);
}

__global__ void deg_to_norm_kernel(float* __restrict__ v, int n) {
    int i = blockIdx.x * blockDim.x + threadIdx.x;
    if (i >= n) return;
    float d = v[i];
    v[i] = (d > 0.0f) ? rsqrtf(d) : 1.0f;
}

// ---------------- generic FP32 WMMA GEMM: C = act(A[M,K] @ W[K,HID] + bias) ----------------
// one wave per 16x16 output tile; A/B fragments per V_WMMA_F32_16X16X4_F32 lane layout
template<int K, bool RELU>
__global__ void gemm_wmma_k(const float* __restrict__ A,
                            const float* __restrict__ W,
                            const float* __restrict__ bias,
                            float* __restrict__ C) {
    const int lane = threadIdx.x;      // 0..31
    const int hf   = lane >> 4;        // 0: K{0,1}, 1: K{2,3}
    const int l    = lane & 15;
    const int row  = blockIdx.x * 16 + l;   // A row for this lane's fragment
    const int col  = blockIdx.y * 16 + l;   // B/C column for this lane

    const float* Arow = A + (size_t)row * K + hf * 2;
    const float* Wcol = W + (size_t)(hf * 2) * HID + col;

    v8f c = {};
#pragma unroll
    for (int k = 0; k < K; k += 4) {
        v2f a, b;
        a.x = Arow[k];
        a.y = Arow[k + 1];
        b.x = Wcol[(size_t)k * HID];
        b.y = Wcol[(size_t)(k + 1) * HID];
        c = __builtin_amdgcn_wmma_f32_16x16x4_f32(false, a, false, b,
                                                  (short)0, c, false, false);
    }

    const float bv    = bias[col];
    const int   rbase = blockIdx.x * 16 + hf * 8;
#pragma unroll
    for (int i = 0; i < 8; i++) {
        float v = c[i] + bv;
        if (RELU) v = fmaxf(v, 0.0f);
        C[(size_t)(rbase + i) * HID + blockIdx.y * 16 + l] = v;
    }
}

// ---------------- RGCN GEMM: C = act( sum_r (agg[r] * norm_d[r][row]) @ W[r] + b[r] ) ----------------
template<bool RELU>
__global__ void rgcn_gemm_wmma(const float* __restrict__ agg,    // [RR][NN][HID]
                               const float* __restrict__ normd,  // [RR][NN]
                               const float* __restrict__ Wr,     // [RR][HID][HID]
                               const float* __restrict__ br,     // [RR][HID]
                               float* __restrict__ C) {
    const int lane = threadIdx.x;
    const int hf   = lane >> 4;
    const int l    = lane & 15;
    const int row  = blockIdx.x * 16 + l;
    const int col  = blockIdx.y * 16 + l;

    v8f   c    = {};
    float bsum = 0.0f;
    for (int r = 0; r < RR; r++) {
        const float  s    = normd[(size_t)r * NN + row];
        const float* Arow = agg + ((size_t)r * NN + row) * HID + hf * 2;
        const float* Wcol = Wr + (size_t)r * HID * HID + (size_t)(hf * 2) * HID + col;
#pragma unroll 8
        for (int k = 0; k < HID; k += 4) {
            v2f a, b;
            a.x = Arow[k] * s;
            a.y = Arow[k + 1] * s;
            b.x = Wcol[(size_t)k * HID];
            b.y = Wcol[(size_t)(k + 1) * HID];
            c = __builtin_amdgcn_wmma_f32_16x16x4_f32(false, a, false, b,
                                                      (short)0, c, false, false);
        }
        bsum += br[r * HID + col];
    }

    const int rbase = blockIdx.x * 16 + hf * 8;
#pragma unroll
    for (int i = 0; i < 8; i++) {
        float v = c[i] + bsum;
        if (RELU) v = fmaxf(v, 0.0f);
        C[(size_t)(rbase + i) * HID + blockIdx.y * 16 + l] = v;
    }
}

// ---------------- edge scatter: agg[r][dst] += h[src] * norm_s[r][src] ----------------
// one wave per edge; each lane moves 4 channels (float4 load, 4 f32 atomics)
__global__ void scatter_msg_kernel(const float* __restrict__ h,
                                   const int* __restrict__ src,
                                   const int* __restrict__ dst,
                                   const float* __restrict__ norm_s,
                                   float* __restrict__ agg) {
    long long eid = (long long)blockIdx.x * 8 + (threadIdx.x >> 5);
    if (eid >= (long long)RR * EE) return;
    int r    = (int)(eid / EE);
    int lane = threadIdx.x & 31;
    int s    = src[eid];
    int d    = dst[eid];
    float ns = norm_s[(size_t)r * NN + s];
    const float4* hp = (const float4*)(h + (size_t)s * HID);
    float4 v = hp[lane];
    float* out = agg + ((size_t)r * NN + d) * HID + lane * 4;
    atomicAdd(out + 0, v.x * ns);
    atomicAdd(out + 1, v.y * ns);
    atomicAdd(out + 2, v.z * ns);
    atomicAdd(out + 3, v.w * ns);
}

// ---------------- ragged gather + masked mean ----------------
__global__ void row_mean_kernel(const float* __restrict__ h,
                                const int* __restrict__ row_idx,
                                const int* __restrict__ row_mask,
                                float* __restrict__ out) {
    int row = blockIdx.x;
    int cch = threadIdx.x;  // 0..127
    float sum = 0.0f;
    int   cnt = 0;
#pragma unroll
    for (int f = 0; f < FF; f++) {
        int m   = row_mask[row * FF + f];
        int idx = row_idx[row * FF + f];
        if (m) { sum += h[(size_t)idx * HID + cch]; cnt++; }
    }
    float denom = (cnt > 0) ? (float)cnt : 1.0f;
    out[(size_t)row * HID + cch] = sum / denom;
}

// ---------------- final head (N=10): trivial VALU GEMM ----------------
__global__ void head_gemm10_kernel(const float* __restrict__ X,
                                   const float* __restrict__ W,
                                   const float* __restrict__ b,
                                   float* __restrict__ out) {
    int i = blockIdx.x * blockDim.x + threadIdx.x;
    if (i >= NROWS * NCLS) return;
    int row = i / NCLS, cc = i % NCLS;
    const float* x = X + (size_t)row * HID;
    float s = b[cc];
#pragma unroll 8
    for (int k = 0; k < HID; k++) s += x[k] * W[k * NCLS + cc];
    out[i] = s;
}

extern "C" void kernel_launch(void* const* d_in, const int* in_sizes, int n_in,
                              void* d_out, int out_size, void* d_ws, size_t ws_size,
                              hipStream_t stream) {
    const float* node_feats = (const float*)d_in[0];
    const int*   edges_src  = (const int*)d_in[1];
    const int*   edges_dst  = (const int*)d_in[2];
    const int*   row_idx    = (const int*)d_in[3];
    const int*   row_mask   = (const int*)d_in[4];
    const float* W_in = (const float*)d_in[5];
    const float* b_in = (const float*)d_in[6];
    const float* W1   = (const float*)d_in[7];
    const float* b1   = (const float*)d_in[8];
    const float* W2   = (const float*)d_in[9];
    const float* b2   = (const float*)d_in[10];
    const float* Wm1  = (const float*)d_in[11];
    const float* bm1  = (const float*)d_in[12];
    const float* Wm2  = (const float*)d_in[13];
    const float* bm2  = (const float*)d_in[14];
    const float* Wm3  = (const float*)d_in[15];
    const float* bm3  = (const float*)d_in[16];
    float* out = (float*)d_out;

    // workspace carve (floats)
    float* ws     = (float*)d_ws;
    float* norm_s = ws;                            // [RR][NN]   (deg first, norm after)
    float* norm_d = norm_s + (size_t)RR * NN;      // [RR][NN]
    float* h0     = norm_d + (size_t)RR * NN;      // [NN][HID]
    float* h1     = h0 + (size_t)NN * HID;         // [NN][HID]
    float* agg    = h1 + (size_t)NN * HID;         // [RR][NN][HID]
    float* mlp_a  = agg;                           // alias: free after layer GEMMs
    float* mlp_b  = agg + (size_t)NROWS * HID;

    // degrees -> norms (both deg buffers are contiguous)
    hipMemsetAsync(norm_s, 0, (size_t)2 * RR * NN * sizeof(float), stream);
    {
        int n = RR * EE;
        count_deg_kernel<<<(n + 255) / 256, 256, 0, stream>>>(edges_src, edges_dst, norm_s, norm_d);
        int m = 2 * RR * NN;
        deg_to_norm_kernel<<<(m + 255) / 256, 256, 0, stream>>>(norm_s, m);
    }

    // input linear + relu
    gemm_wmma_k<INDIM, true><<<dim3(NN / 16, HID / 16), 32, 0, stream>>>(node_feats, W_in, b_in, h0);

    long long ne = (long long)RR * EE;
    int sblocks  = (int)((ne + 7) / 8);

    // RGCN layer 1
    hipMemsetAsync(agg, 0, (size_t)RR * NN * HID * sizeof(float), stream);
    scatter_msg_kernel<<<sblocks, 256, 0, stream>>>(h0, edges_src, edges_dst, norm_s, agg);
    rgcn_gemm_wmma<true><<<dim3(NN / 16, HID / 16), 32, 0, stream>>>(agg, norm_d, W1, b1, h1);

    // RGCN layer 2
    hipMemsetAsync(agg, 0, (size_t)RR * NN * HID * sizeof(float), stream);
    scatter_msg_kernel<<<sblocks, 256, 0, stream>>>(h1, edges_src, edges_dst, norm_s, agg);
    rgcn_gemm_wmma<false><<<dim3(NN / 16, HID / 16), 32, 0, stream>>>(agg, norm_d, W2, b2, h0);

    // ragged gather + masked mean
    row_mean_kernel<<<NROWS, HID, 0, stream>>>(h0, row_idx, row_mask, mlp_a);

    // MLP head
    gemm_wmma_k<HID, true><<<dim3(NROWS / 16, HID / 16), 32, 0, stream>>>(mlp_a, Wm1, bm1, mlp_b);
    gemm_wmma_k<HID, true><<<dim3(NROWS / 16, HID / 16), 32, 0, stream>>>(mlp_b, Wm2, bm2, mlp_a);
    {
        int n = NROWS * NCLS;
        head_gemm10_kernel<<<(n + 255) / 256, 256, 0, stream>>>(mlp_a, Wm3, bm3, out);
    }
}